// Laplacian_28217935135197
// MI455X (gfx1250) — compile-verified
//
#include <hip/hip_runtime.h>
#include <math.h>

// ---------------------------------------------------------------------------
// Cotangent-Laplacian apply:  out = Lap(V, faces) @ V, computed sparsely.
// Per face (i0,i1,i2) with half-cot weights c0,c1,c2 (reference's C values):
//   out[i0] += c2*(v1-v2) + c1*(v1-v3)
//   out[i1] += c0*(v2-v3) + c2*(v2-v1)
//   out[i2] += c1*(v3-v1) + c0*(v3-v2)
// V[b] (82.7 KB) is staged into LDS (320 KB/WGP on CDNA5) with the async
// global->LDS data mover; gathers then read LDS, scatters use f32 atomics.
// ---------------------------------------------------------------------------

#define AS1 __attribute__((address_space(1)))
#define AS3 __attribute__((address_space(3)))

typedef int v4i __attribute__((vector_size(4 * sizeof(int))));

#ifndef __has_builtin
#define __has_builtin(x) 0
#endif

#if __has_builtin(__builtin_amdgcn_global_load_async_to_lds_b128) && \
    __has_builtin(__builtin_amdgcn_global_load_async_to_lds_b32)
#define HAVE_ASYNC_LDS 1
#else
#define HAVE_ASYNC_LDS 0
#endif

namespace {
constexpr int kB  = 2;
constexpr int kVN = 6890;
constexpr int kFN = 13776;
constexpr int kVN3 = kVN * 3;          // 20670 floats = 82680 bytes of LDS
constexpr int kThreads = 256;          // 8 x wave32
constexpr int kFacesPerThread = 4;
constexpr int kWGsPerBatch =
    (kFN + kThreads * kFacesPerThread - 1) / (kThreads * kFacesPerThread);  // 14
}  // namespace

__device__ __forceinline__ void atomic_add_rlx(float* p, float v) {
  // relaxed, device-scope fp32 atomic -> global_atomic_add_f32
  __hip_atomic_fetch_add(p, v, __ATOMIC_RELAXED, __HIP_MEMORY_SCOPE_AGENT);
}

__global__ __launch_bounds__(kThreads)
void lap_zero_kernel(float* __restrict__ out, int n) {
  const int i = blockIdx.x * blockDim.x + threadIdx.x;
  if (i < n) out[i] = 0.0f;
}

__global__ __launch_bounds__(kThreads)
void cot_laplacian_kernel(const float* __restrict__ V,
                          const int* __restrict__ faces,
                          float* __restrict__ out) {
  __shared__ float sV[kVN3];

  const int b = blockIdx.y;
  const float* gV = V + (size_t)b * kVN3;

  // ---- Stage V[b] into LDS via CDNA5 async global->LDS loads -------------
#if HAVE_ASYNC_LDS
  constexpr int kChunks = kVN3 / 4;                 // 5167 x 16B chunks
  for (int c = threadIdx.x; c < kChunks; c += kThreads) {
    __builtin_amdgcn_global_load_async_to_lds_b128(
        (AS1 v4i*)(gV + (size_t)c * 4), (AS3 v4i*)(sV + c * 4), 0, 0);
  }
  constexpr int kTail = kVN3 - kChunks * 4;         // 2 dwords
  if ((int)threadIdx.x < kTail) {
    __builtin_amdgcn_global_load_async_to_lds_b32(
        (AS1 int*)(gV + kChunks * 4 + threadIdx.x),
        (AS3 int*)(sV + kChunks * 4 + threadIdx.x), 0, 0);
  }
#if __has_builtin(__builtin_amdgcn_s_wait_asynccnt)
  __builtin_amdgcn_s_wait_asynccnt(0);
#else
  asm volatile("s_wait_asynccnt 0" ::: "memory");
#endif
#else
  for (int i = threadIdx.x; i < kVN3; i += kThreads) sV[i] = gV[i];
#endif
  __syncthreads();

  // ---- Per-face cotangent weights + scatter ------------------------------
  const int base = blockIdx.x * (kThreads * kFacesPerThread);
  for (int k = 0; k < kFacesPerThread; ++k) {
    const int f = base + k * kThreads + (int)threadIdx.x;
    if (f >= kFN) break;

    const int* fp = faces + ((size_t)b * kFN + f) * 3;
    const int i0 = fp[0], i1 = fp[1], i2 = fp[2];

    const float v1x = sV[i0 * 3 + 0], v1y = sV[i0 * 3 + 1], v1z = sV[i0 * 3 + 2];
    const float v2x = sV[i1 * 3 + 0], v2y = sV[i1 * 3 + 1], v2z = sV[i1 * 3 + 2];
    const float v3x = sV[i2 * 3 + 0], v3y = sV[i2 * 3 + 1], v3z = sV[i2 * 3 + 2];

    const float e1x = v2x - v3x, e1y = v2y - v3y, e1z = v2z - v3z;  // v2-v3
    const float e2x = v3x - v1x, e2y = v3y - v1y, e2z = v3z - v1z;  // v3-v1
    const float e3x = v1x - v2x, e3y = v1y - v2y, e3z = v1z - v2z;  // v1-v2

    const float l1s = e1x * e1x + e1y * e1y + e1z * e1z;
    const float l2s = e2x * e2x + e2y * e2y + e2z * e2z;
    const float l3s = e3x * e3x + e3y * e3y + e3z * e3z;
    const float l1 = sqrtf(l1s), l2 = sqrtf(l2s), l3 = sqrtf(l3s);

    const float sp = 0.5f * (l1 + l2 + l3);
    const float A  = 2.0f * sqrtf(sp * (sp - l1) * (sp - l2) * (sp - l3));
    const float inv = 1.0f / (4.0f * A);
    const float c0 = (l2s + l3s - l1s) * inv;   // cot23  -> edge (i1,i2)
    const float c1 = (l1s + l3s - l2s) * inv;   // cot31  -> edge (i2,i0)
    const float c2 = (l1s + l2s - l3s) * inv;   // cot12  -> edge (i0,i1)

    float* o0 = out + ((size_t)b * kVN + i0) * 3;
    float* o1 = out + ((size_t)b * kVN + i1) * 3;
    float* o2 = out + ((size_t)b * kVN + i2) * 3;

    // combined per-vertex contributions (9 atomics/face instead of 18)
    atomic_add_rlx(o0 + 0, c2 * e3x - c1 * e2x);
    atomic_add_rlx(o0 + 1, c2 * e3y - c1 * e2y);
    atomic_add_rlx(o0 + 2, c2 * e3z - c1 * e2z);

    atomic_add_rlx(o1 + 0, c0 * e1x - c2 * e3x);
    atomic_add_rlx(o1 + 1, c0 * e1y - c2 * e3y);
    atomic_add_rlx(o1 + 2, c0 * e1z - c2 * e3z);

    atomic_add_rlx(o2 + 0, c1 * e2x - c0 * e1x);
    atomic_add_rlx(o2 + 1, c1 * e2y - c0 * e1y);
    atomic_add_rlx(o2 + 2, c1 * e2z - c0 * e1z);
  }
}

extern "C" void kernel_launch(void* const* d_in, const int* in_sizes, int n_in,
                              void* d_out, int out_size, void* d_ws, size_t ws_size,
                              hipStream_t stream) {
  (void)in_sizes; (void)n_in; (void)d_ws; (void)ws_size;

  const float* V     = (const float*)d_in[0];   // (B, VN, 3) fp32
  const int*   faces = (const int*)d_in[1];     // (B, FN, 3) int32
  float*       out   = (float*)d_out;           // (B, VN, 3) fp32

  const int n = kB * kVN3;                      // == out_size (41340)
  lap_zero_kernel<<<(n + kThreads - 1) / kThreads, kThreads, 0, stream>>>(out, n);

  dim3 grid(kWGsPerBatch, kB);
  cot_laplacian_kernel<<<grid, kThreads, 0, stream>>>(V, faces, out);
  (void)out_size;
}